// GCNEncoder_12043088298540
// MI455X (gfx1250) — compile-verified
//
#include <hip/hip_runtime.h>

typedef __attribute__((ext_vector_type(2))) float v2f;
typedef __attribute__((ext_vector_type(8))) float v8f;

#define HAS_ASYNC_LDS __has_builtin(__builtin_amdgcn_global_load_async_to_lds_b128)

#if HAS_ASYNC_LDS
// b128 async-copy operand types: pointer to 16B int vector in AS(1)/AS(3)
typedef int v4i_b128 __attribute__((vector_size(16)));
typedef __attribute__((address_space(1))) v4i_b128* g_b128_ptr;
typedef __attribute__((address_space(3))) v4i_b128* l_b128_ptr;
#endif

// ---------------------------------------------------------------- utilities

__global__ void zero_f32_kernel(float* __restrict__ p, long long n) {
    long long i = (long long)blockIdx.x * blockDim.x + threadIdx.x;
    long long stride = (long long)gridDim.x * blockDim.x;
    for (; i < n; i += stride) p[i] = 0.0f;
}

__global__ void degree_kernel(const long long* __restrict__ src,
                              const long long* __restrict__ dst,
                              float* __restrict__ outdeg,
                              float* __restrict__ indeg, int E) {
    int e = blockIdx.x * blockDim.x + threadIdx.x;
    if (e < E) {
        atomicAdd(&outdeg[(int)src[e]], 1.0f);
        atomicAdd(&indeg[(int)dst[e]], 1.0f);
    }
}

// deg -> deg^{-1/2} with zero-degree clamped to 1 (dgl norm='both')
__global__ void norm_kernel(float* __restrict__ d, int n) {
    int i = blockIdx.x * blockDim.x + threadIdx.x;
    if (i < n) {
        float v = d[i];
        d[i] = (v > 0.0f) ? rsqrtf(v) : 1.0f;
    }
}

// ------------------------------------------------- edge scatter (aggregate)
// One wave per edge, 128 features per row, 4 consecutive floats per lane.
// agg[dst[e]] += feat[src[e]] * (scale ? scale[src[e]] : 1)
__global__ void scatter_kernel(const float* __restrict__ feat,
                               const long long* __restrict__ src,
                               const long long* __restrict__ dst,
                               const float* __restrict__ scale,
                               float* __restrict__ agg, int E) {
    const int wave = threadIdx.x >> 5;
    const int lane = threadIdx.x & 31;
    const int e = blockIdx.x * (blockDim.x >> 5) + wave;
    if (e >= E) return;
    const long long s = src[e];
    const long long d = dst[e];
    const float sc = scale ? scale[s] : 1.0f;
    const float4 v = *(const float4*)(feat + s * 128 + lane * 4);
    float* ap = agg + d * 128 + lane * 4;
    atomicAdd(ap + 0, v.x * sc);
    atomicAdd(ap + 1, v.y * sc);
    atomicAdd(ap + 2, v.z * sc);
    atomicAdd(ap + 3, v.w * sc);
}

// -------------------------------------------------------------- WMMA GEMM
// out[M x N] = act( rowscale[m] * (A @ W)[m,n] + bias[n] )
// Block: 256 threads = 8 waves. Block covers rows [16*bx,16*bx+16),
// cols [128*by, 128*by+128); wave w owns 16x16 tile at col 128*by+16*w.
// A tile (16 x K) staged in LDS via CDNA5 async global->LDS copies when
// the toolchain exposes them (ASYNCcnt path), else via VGPR staging.
#define LPAD 4   // 4 floats -> padded row stride (K+4)*4B is 16B-aligned

template <int K, int N>
__global__ __launch_bounds__(256) void gemm_kernel(
        const float* __restrict__ A,        // [M x K]
        const float* __restrict__ W,        // [K x N]
        const float* __restrict__ bias,     // [N] or nullptr
        const float* __restrict__ rowscale, // [M] or nullptr
        float* __restrict__ out,            // [M x N]
        int do_relu) {
    __shared__ float As[16 * (K + LPAD)];

    const int tid  = threadIdx.x;
    const int wave = tid >> 5;
    const int lane = tid & 31;
    const int m0   = blockIdx.x * 16;
    const int col  = blockIdx.y * 128 + wave * 16 + (lane & 15);

    // ---- stage the 16 x K A-tile into LDS ----
#if HAS_ASYNC_LDS
    // 16-byte chunks per lane; row stride (K+LPAD) keeps chunks 16B-aligned.
    #pragma unroll
    for (int i = tid * 4; i < 16 * K; i += 256 * 4) {
        const int r = i / K;
        const int c = i - r * K;
        __builtin_amdgcn_global_load_async_to_lds_b128(
            (g_b128_ptr)(void*)(A + (long long)(m0 + r) * K + c),
            (l_b128_ptr)(void*)&As[r * (K + LPAD) + c],
            /*offset=*/0, /*cpol=*/0);
    }
  #if __has_builtin(__builtin_amdgcn_s_wait_asynccnt)
    __builtin_amdgcn_s_wait_asynccnt(0);
  #else
    asm volatile("s_wait_asynccnt 0" ::: "memory");
  #endif
#else
    #pragma unroll 4
    for (int i = tid; i < 16 * K; i += 256) {
        const int r = i / K;
        const int c = i - r * K;
        As[r * (K + LPAD) + c] = A[(long long)(m0 + r) * K + c];
    }
#endif
    __syncthreads();

    // fragment addressing per CDNA5 WMMA VGPR layout:
    // lanes 0-15 cover K pair {k0,k0+1}, lanes 16-31 cover {k0+2,k0+3}
    const int row  = lane & 15;
    const int kb   = (lane >> 4) * 2;
    const int half = lane >> 4;

    v8f acc = {};
    #pragma unroll 8
    for (int k0 = 0; k0 < K; k0 += 4) {
        const v2f a = *(const v2f*)(&As[row * (K + LPAD) + k0 + kb]);
        v2f b;
        b.x = W[(long long)(k0 + kb) * N + col];
        b.y = W[(long long)(k0 + kb + 1) * N + col];
        acc = __builtin_amdgcn_wmma_f32_16x16x4_f32(
                /*neg_a=*/false, a, /*neg_b=*/false, b,
                /*c_mod=*/(short)0, acc, /*reuse_a=*/false, /*reuse_b=*/false);
    }

    const float bv = bias ? bias[col] : 0.0f;
    #pragma unroll
    for (int v = 0; v < 8; v++) {
        const int m = m0 + v + 8 * half;
        const float rs = rowscale ? rowscale[m] : 1.0f;
        float val = acc[v] * rs + bv;
        if (do_relu) val = fmaxf(val, 0.0f);
        out[(long long)m * N + col] = val;
    }
}

// ----------------------------------------------------------- final epilogue
// out[m,c] = relu(out[m,c] * inn[m] + bias[c]),  F == 128
__global__ void finalize_kernel(float* __restrict__ out,
                                const float* __restrict__ inn,
                                const float* __restrict__ bias,
                                long long total) {
    long long i = (long long)blockIdx.x * blockDim.x + threadIdx.x;
    if (i < total) {
        const int m = (int)(i >> 7);
        const int c = (int)(i & 127);
        out[i] = fmaxf(out[i] * inn[m] + bias[c], 0.0f);
    }
}

// ------------------------------------------------------------------ driver

extern "C" void kernel_launch(void* const* d_in, const int* in_sizes, int n_in,
                              void* d_out, int out_size, void* d_ws, size_t ws_size,
                              hipStream_t stream) {
    const float* x        = (const float*)d_in[0];
    const float* W1       = (const float*)d_in[1];
    const float* b1       = (const float*)d_in[2];
    const float* W2       = (const float*)d_in[3];
    const float* b2       = (const float*)d_in[4];
    const long long* src  = (const long long*)d_in[5];
    const long long* dst  = (const long long*)d_in[6];

    const int HID  = in_sizes[2];            // 256
    const int INF  = in_sizes[1] / HID;      // 128
    const int OUTF = in_sizes[4];            // 128
    const int Nn   = in_sizes[0] / INF;      // 100000
    const int E    = in_sizes[5];            // 1600000

    float* base = (float*)d_ws;
    float* outn = base;                          // [Nn]   degree -> out_norm
    float* inn  = base + Nn;                     // [Nn]   degree -> in_norm
    float* buf  = inn + Nn;                      // [Nn x 128]  agg1, then h1@W2
    float* h1   = buf + (size_t)Nn * INF;        // [Nn x 256]
    float* out  = (float*)d_out;

    // 1) degrees -> norms (outn/inn are contiguous: one pass each)
    zero_f32_kernel<<<1024, 256, 0, stream>>>(outn, 2LL * Nn);
    degree_kernel<<<(E + 255) / 256, 256, 0, stream>>>(src, dst, outn, inn, E);
    norm_kernel<<<(2 * Nn + 255) / 256, 256, 0, stream>>>(outn, 2 * Nn);

    // 2) layer-1 aggregation: buf[dst] += x[src] * outn[src]
    zero_f32_kernel<<<2048, 256, 0, stream>>>(buf, (long long)Nn * INF);
    scatter_kernel<<<(E + 7) / 8, 256, 0, stream>>>(x, src, dst, outn, buf, E);

    // 3) h1 = relu( inn .* (buf @ W1) + b1 )        [Nn x 256]
    gemm_kernel<128, 256><<<dim3(Nn / 16, 2), 256, 0, stream>>>(
        buf, W1, b1, inn, h1, /*relu=*/1);

    // 4) buf = outn .* (h1 @ W2)                    [Nn x 128]
    gemm_kernel<256, 128><<<dim3(Nn / 16, 1), 256, 0, stream>>>(
        h1, W2, nullptr, outn, buf, /*relu=*/0);

    // 5) layer-2 aggregation directly into d_out
    zero_f32_kernel<<<2048, 256, 0, stream>>>(out, (long long)Nn * OUTF);
    scatter_kernel<<<(E + 7) / 8, 256, 0, stream>>>(buf, src, dst, nullptr, out, E);

    // 6) out = relu(out .* inn + b2)
    const long long total = (long long)Nn * OUTF;
    finalize_kernel<<<(int)((total + 255) / 256), 256, 0, stream>>>(out, inn, b2, total);
}